// CRFDecoder_34694745817301
// MI455X (gfx1250) — compile-verified
//
#include <hip/hip_runtime.h>

// CRF Viterbi decode for gfx1250 (MI455X), one wave32 per sentence.
// scores[p,n] = fv[p] + trans[p,n] is built with V_WMMA_F32_16X16X4_F32 as
// a rank-1 outer-sum:  D = A(fv,K=0) x B(ones,K=0) + C(trans tile).
// The max/argmax (tropical part) stays on the VALU and co-executes with the
// XDL WMMA pipe. Emissions stream in via global_load_async_to_lds_b32 +
// s_wait_asynccnt. Backpointers (64KB/sentence) live entirely in LDS.
// Wave-private recurrence => no barriers at all in the T-loop.

typedef __attribute__((ext_vector_type(2))) float v2f;
typedef __attribute__((ext_vector_type(8))) float v8f;

#define LBL   64
#define BATCH 256
#define TLEN  1024
#define BOSL  62
#define EOSL  63
#define NEGV  (-10000.0f)
#define LOOK  7     // async pipeline depth (steps)
#define RING  8     // emission ring slots (> LOOK)

__global__ __launch_bounds__(32) void crf_viterbi_wmma(
    const float* __restrict__ X,       // [B, T, L]
    const float* __restrict__ trans,   // [L, L]
    float* __restrict__ out)           // [B] scores ++ [B, T] paths (as float)
{
  __shared__ __align__(16) float fvbuf[LBL];        // viterbi vars (wave-private use)
  __shared__ __align__(16) float emis[RING][LBL];   // async emission ring
  __shared__ __align__(16) float termbuf[LBL];      // termination scores
  __shared__ unsigned char  bparr[TLEN * LBL];      // backpointers, 64 KB

  const int l  = threadIdx.x;        // lane 0..31
  const int b  = blockIdx.x;         // sentence
  const int lo = l & 15;             // N (and M) position within a 16x16 tile
  const int hi = (l >> 4) & 1;       // lane half: holds M = r + 8*hi of each tile
  const int n1 = hi * 32 + lo;       // this lane owns columns n1 and n1+16
  const int n2 = n1 + 16;

  // ---- Resident C tiles: c[Pt][Nt][r] = trans[Pt*16 + r + 8*hi][Nt*16 + lo]
  // (exactly the 16x16 f32 C/D VGPR layout: VGPR r -> M=r / M=r+8)
  v8f c[4][4];
#pragma unroll
  for (int Pt = 0; Pt < 4; ++Pt)
#pragma unroll
    for (int Nt = 0; Nt < 4; ++Nt)
#pragma unroll
      for (int r = 0; r < 8; ++r)
        c[Pt][Nt][r] = trans[(Pt * 16 + r + 8 * hi) * LBL + Nt * 16 + lo];

  const float teos1 = trans[n1 * LBL + EOSL];
  const float teos2 = trans[n2 * LBL + EOSL];

  // B matrix (4x16): only row K=0 is ones -> VGPR0 lanes 0-15; rest zero.
  v2f bm;
  bm.x = (hi == 0) ? 1.0f : 0.0f;
  bm.y = 0.0f;

  // init[BOS]=0 else NEG
  fvbuf[n1] = (n1 == BOSL) ? 0.0f : NEGV;
  fvbuf[n2] = (n2 == BOSL) ? 0.0f : NEGV;

  // ---- Async emission pipeline: this wave loads all 64 columns per step
  // (lane l loads columns l and l+32 -> two coalesced 128B segments).
  const float* Xb = X + (size_t)b * TLEN * LBL;
  unsigned ldsEmis = (unsigned)(unsigned long long)&emis[0][0];  // LDS byte offset
  for (int k = 0; k < LOOK; ++k) {
    unsigned g0 = (unsigned)((k * LBL + l) * 4);
    unsigned s0 = ldsEmis + (unsigned)(((k & (RING - 1)) * LBL + l) * 4);
    asm volatile("global_load_async_to_lds_b32 %0, %1, %2"
                 :: "v"(s0), "v"(g0), "s"(Xb) : "memory");
    asm volatile("global_load_async_to_lds_b32 %0, %1, %2"
                 :: "v"(s0 + 128u), "v"(g0 + 128u), "s"(Xb) : "memory");
  }

  float fv1 = NEGV, fv2 = NEGV;

  for (int t = 0; t < TLEN; ++t) {
    // Prefetch step t+LOOK (source clamped near the tail so the in-flight
    // count stays uniform and a constant s_wait works).
    {
      int tl = t + LOOK;
      int tc = (tl < TLEN) ? tl : (TLEN - 1);
      unsigned g0 = (unsigned)((tc * LBL + l) * 4);
      unsigned s0 = ldsEmis + (unsigned)(((tl & (RING - 1)) * LBL + l) * 4);
      asm volatile("global_load_async_to_lds_b32 %0, %1, %2"
                   :: "v"(s0), "v"(g0), "s"(Xb) : "memory");
      asm volatile("global_load_async_to_lds_b32 %0, %1, %2"
                   :: "v"(s0 + 128u), "v"(g0 + 128u), "s"(Xb) : "memory");
    }
    // Outstanding after issue = 2 loads x steps t..t+7 = 16; waiting <=14
    // guarantees both loads of step t have landed in LDS.
    asm volatile("s_wait_asynccnt 14" ::: "memory");

    // Build A tiles (16x4 f32 layout: VGPR0 lanes0-15 = K=0 = fv, rest 0).
    // Same-wave LDS ops are in-order, so last step's fv stores are visible.
    v2f a[4];
#pragma unroll
    for (int Pt = 0; Pt < 4; ++Pt) {
      float f = fvbuf[Pt * 16 + lo];
      a[Pt].x = (hi == 0) ? f : 0.0f;
      a[Pt].y = 0.0f;
    }

    // For each N tile: 4 WMMAs build scores, VALU reduces max/argmax over p.
    float sNt[4];
    int   pNt[4];
#pragma unroll
    for (int Nt = 0; Nt < 4; ++Nt) {
      float sBest = 0.0f;
      int   pBest = 0;
#pragma unroll
      for (int Pt = 0; Pt < 4; ++Pt) {
        // D[m,j] = fv[Pt*16+m] + trans[Pt*16+m][Nt*16+j]
        v8f d = __builtin_amdgcn_wmma_f32_16x16x4_f32(
            false, a[Pt], false, bm, (short)0, c[Pt][Nt], false, false);

        // In-lane reduce over r (ascending p, '>=' keep-left => first-max).
        float vv[8]; int ii[8];
#pragma unroll
        for (int r = 0; r < 8; ++r) { vv[r] = d[r]; ii[r] = Pt * 16 + r; }
#pragma unroll
        for (int len = 8; len > 1; len >>= 1) {
#pragma unroll
          for (int i = 0; i < (len >> 1); ++i) {
            bool kp = vv[2 * i] >= vv[2 * i + 1];
            vv[i] = kp ? vv[2 * i] : vv[2 * i + 1];
            ii[i] = kp ? ii[2 * i] : ii[2 * i + 1];
          }
        }
        if (Pt == 0) { sBest = vv[0]; pBest = ii[0]; }
        else {
          bool kp = sBest >= vv[0];   // left tile has smaller p => keep-left
          sBest = kp ? sBest : vv[0];
          pBest = kp ? pBest : ii[0];
        }
      }
      sNt[Nt] = sBest;
      pNt[Nt] = pBest;
    }

    // Cross-half combine: other half of the column lives in lane l^16
    // (M offset +8). Compare on global p for exact first-occurrence ties.
#pragma unroll
    for (int Nt = 0; Nt < 4; ++Nt) {
      int   pG = pNt[Nt] + 8 * hi;                 // local -> global p
      float sO = __shfl_xor(sNt[Nt], 16, 32);
      int   pO = __shfl_xor(pG, 16, 32);
      bool  kp = (sNt[Nt] > sO) || ((sNt[Nt] == sO) && (pG < pO));
      sNt[Nt] = kp ? sNt[Nt] : sO;
      pNt[Nt] = kp ? pG : pO;
    }

    // This lane owns columns n1 (tile 2*hi) and n2 (tile 2*hi+1).
    float sA = hi ? sNt[2] : sNt[0];
    int   pA = hi ? pNt[2] : pNt[0];
    float sB = hi ? sNt[3] : sNt[1];
    int   pB = hi ? pNt[3] : pNt[1];

    float e1 = emis[t & (RING - 1)][n1];
    float e2 = emis[t & (RING - 1)][n2];
    fv1 = sA + e1;
    fv2 = sB + e2;
    fvbuf[n1] = fv1;
    fvbuf[n2] = fv2;
    bparr[t * LBL + n1] = (unsigned char)pA;
    bparr[t * LBL + n2] = (unsigned char)pB;
    // No barrier: single wave, LDS ops in program order.
  }

  // ---- Termination + backtrace, all from LDS ----
  termbuf[n1] = fv1 + teos1;
  termbuf[n2] = fv2 + teos2;
  __syncthreads();   // single-wave WG: hardware NOP, keeps compiler ordering

  if (l == 0) {
    float bestS = termbuf[0];
    int   last  = 0;
    for (int p = 1; p < LBL; ++p) {
      float s = termbuf[p];
      if (s > bestS) { bestS = s; last = p; }   // strict '>' => first max
    }
    out[b] = bestS;

    float* paths = out + BATCH;
    int curTag = last;
    for (int t = TLEN - 1; t >= 0; --t) {
      paths[(size_t)b * TLEN + t] = (float)curTag;  // y[t] = carry
      curTag = bparr[t * LBL + curTag];             // carry = bp[t][y[t]]
    }
  }
}

extern "C" void kernel_launch(void* const* d_in, const int* in_sizes, int n_in,
                              void* d_out, int out_size, void* d_ws, size_t ws_size,
                              hipStream_t stream) {
  (void)in_sizes; (void)n_in; (void)d_ws; (void)ws_size; (void)out_size;
  const float* X     = (const float*)d_in[0];   // [256,1024,64] f32
  const float* trans = (const float*)d_in[1];   // [64,64] f32
  float* out = (float*)d_out;                   // 256 scores ++ 256*1024 tags
  crf_viterbi_wmma<<<dim3(BATCH), dim3(32), 0, stream>>>(X, trans, out);
}